// Attention_33741263077380
// MI455X (gfx1250) — compile-verified
//
#include <hip/hip_runtime.h>

// Problem constants (from reference): B=8, L=2048, D=1024, K=1024 (=L/2)
#define BB   8
#define LL   2048
#define DD   1024
#define KK   1024

typedef __attribute__((ext_vector_type(2))) float v2f;
typedef __attribute__((ext_vector_type(4))) float v4f;
typedef __attribute__((ext_vector_type(8))) float v8f;

// ---------------------------------------------------------------------------
// Kernel 1: router scores via V_WMMA_F32_16X16X4_F32.
// One wave32 computes scores for a 16-row tile of x.
//   A (16x4 f32, 2 VGPRs): lanes 0-15 carry rows M=0..15 with K={0,1},
//                          lanes 16-31 carry the same rows with K={2,3}.
//   B (4x16 f32, 2 VGPRs): column N=0 = router_w[k..k+3], all other cols 0.
//   C/D (16x16 f32, 8 VGPRs): column N=0 (lanes 0 and 16) = the 16 scores.
// B-fragment is built branchlessly (v_cndmask, not exec masking) so the inner
// loop is a straight load/load/wmma stream with EXEC all-1s throughout.
// ---------------------------------------------------------------------------
__global__ __launch_bounds__(128) void router_scores_wmma(
    const float* __restrict__ x,      // [B*L, D]
    const float* __restrict__ w,      // [D]
    const float* __restrict__ bias,   // [1]
    float* __restrict__ rw)           // [B*L]
{
    const int lane      = threadIdx.x & 31;
    const int waveInBlk = threadIdx.x >> 5;
    const int tile      = blockIdx.x * 4 + waveInBlk;   // 16-row tile id
    const long rowBase  = (long)tile * 16;

    const int n    = lane & 15;          // N column / M row within tile
    const int kOff = (lane >> 4) << 1;   // 0 for lanes 0-15, 2 for lanes 16-31
    const bool col0 = (n == 0);

    const float* xrow = x + (rowBase + n) * (long)DD + kOff;
    const float* wptr = w + kOff;

    v8f c = {};
    for (int k = 0; k < DD; k += 4) {
        // A fragment: two consecutive f32 of this lane's row
        v2f a = *(const v2f*)(xrow + k);

        // B fragment: broadcast-load the weights in all lanes, then select
        // column 0 with v_cndmask (no exec manipulation, no branches).
        v2f wv = *(const v2f*)(wptr + k);
        v2f bf;
        bf.x = col0 ? wv.x : 0.0f;
        bf.y = col0 ? wv.y : 0.0f;

        // 8-arg form: (neg_a, A, neg_b, B, c_mod, C, reuse_a, reuse_b)
        c = __builtin_amdgcn_wmma_f32_16x16x4_f32(
                false, a, false, bf, (short)0, c, false, false);
    }

    // Column N==0 holds the scores: lane 0 -> M=0..7 (VGPR 0..7),
    // lane 16 -> M=8..15 (VGPR 0..7).
    if (col0) {
        const float bb    = bias[0];
        const int   mBase = (lane >> 4) * 8;
#pragma unroll
        for (int v = 0; v < 8; ++v)
            rw[rowBase + mBase + v] = c[v] + bb;
    }
}

// ---------------------------------------------------------------------------
// Kernel 2: exact top-K set per batch via rank selection.
// Element i is in the top-k iff  #{ j : s[j] > s[i]  ||  (s[j]==s[i] && j<i) } < K,
// which reproduces lax.top_k / torch.topk tie-breaking exactly.
// ---------------------------------------------------------------------------
__global__ __launch_bounds__(512) void topk_mask_kernel(
    const float* __restrict__ rw,     // [B*L]
    int* __restrict__ mask)           // [B*L]
{
    __shared__ float s[LL];
    const int b = blockIdx.x;
    const float* r = rw + (long)b * LL;

    for (int i = threadIdx.x; i < LL; i += 512) s[i] = r[i];
    __syncthreads();

    for (int i = threadIdx.x; i < LL; i += 512) {
        const float vi = s[i];
        int rank = 0;
        for (int j = 0; j < LL; ++j) {
            const float vj = s[j];
            rank += (vj > vi) || ((vj == vi) && (j < i));
        }
        mask[(long)b * LL + i] = (rank < KK) ? 1 : 0;
    }
}

// ---------------------------------------------------------------------------
// Kernel 3: scatter-copy.  out[b,i,:] = mask ? x[b, L-1-i, :] : 0.
// One block per output row; b128 loads, non-temporal b128 stores (output is
// write-once, keep it out of L2 so the twice-read x stays resident).
// ---------------------------------------------------------------------------
__global__ __launch_bounds__(256) void scatter_copy_kernel(
    const float* __restrict__ x,      // [B, L, D]
    const int* __restrict__ mask,     // [B*L]
    float* __restrict__ out)          // [B, L, D]
{
    const int row = blockIdx.x;       // 0 .. B*L-1
    const int b   = row >> 11;        // / L
    const int i   = row & (LL - 1);
    const int t   = threadIdx.x;      // 0..255, 4 floats each

    v4f val = {0.f, 0.f, 0.f, 0.f};
    if (mask[row]) {
        const long src = ((long)b * LL + (LL - 1 - i)) * DD;
        val = *(const v4f*)(x + src + t * 4);
    }
    __builtin_nontemporal_store(val, (v4f*)(out + (long)row * DD + t * 4));
}

// ---------------------------------------------------------------------------
extern "C" void kernel_launch(void* const* d_in, const int* in_sizes, int n_in,
                              void* d_out, int out_size, void* d_ws, size_t ws_size,
                              hipStream_t stream) {
    const float* x     = (const float*)d_in[0];   // [B,L,D]
    const float* rwgt  = (const float*)d_in[1];   // [1,D]
    const float* rbias = (const float*)d_in[2];   // [1]
    // d_in[3..5] = wq/wk/wv: dead inputs (attention output is overwritten).

    float* out  = (float*)d_out;
    float* rw   = (float*)d_ws;                                   // B*L floats
    int*   mask = (int*)((char*)d_ws + (size_t)BB * LL * sizeof(float));

    // 1) router scores: B*L/16 = 1024 wave-tiles, 4 waves per 128-thread block
    router_scores_wmma<<<dim3((BB * LL / 16) / 4), dim3(128), 0, stream>>>(
        x, rwgt, rbias, rw);

    // 2) top-k membership mask, one block per batch
    topk_mask_kernel<<<dim3(BB), dim3(512), 0, stream>>>(rw, mask);

    // 3) scatter-copy, one block per output row
    scatter_copy_kernel<<<dim3(BB * LL), dim3(256), 0, stream>>>(x, mask, out);
}